// MSDeformAttn_83648783057213
// MI455X (gfx1250) — compile-verified
//
#include <hip/hip_runtime.h>
#include <math.h>

// Problem constants (from the reference)
#define NTOK   21760
#define BATCH  2
#define CDIM   256
#define NHEADS 8
#define NLEV   4
#define NPTS   4
#define HDIM   32
#define MROWS  (BATCH * NTOK)   // 43520

#define MBLK   4                // 4 x 16 = 64 output rows per wave

typedef float v2f __attribute__((ext_vector_type(2)));
typedef float v8f __attribute__((ext_vector_type(8)));

// ---------------------------------------------------------------------------
// GEMM: out[M,N] = A[M,K] @ W[K,N] + bias[N], fp32 via V_WMMA_F32_16X16X4_F32.
// One wave computes a 64x16 output block (4 M-subtiles sharing one B
// fragment): per k-step -> 4 x b64 A-loads + 2 x b32 B-loads feeding 4 WMMAs
// (1.5 VMEM/WMMA). Fragments are double-buffered: the k+1 loads are issued
// as one batch before the k-step's WMMA group, so load latency is hidden
// under matrix math. All control flow is wave-uniform so EXEC stays all-ones
// across WMMA (ISA requirement). M must be a multiple of 64.
//
// Fragment layouts (CDNA5 ISA 7.12.2, 32-bit operands):
//   A 16x4 : lane l<16  -> M=l,    VGPR0=K0, VGPR1=K1 ; lane l>=16 -> K2,K3
//   B 4x16 : lane l<16  -> N=l,    VGPR0=K0, VGPR1=K1 ; lane l>=16 -> K2,K3
//   D 16x16: VGPR i -> (M = i + 8*(l/16), N = l%16)
// ---------------------------------------------------------------------------
__global__ __launch_bounds__(256)
void gemm_wmma_f32(const float* __restrict__ A,
                   const float* __restrict__ W,
                   const float* __restrict__ bias,
                   float* __restrict__ out,
                   int M, int N, int K) {
    const int lane = threadIdx.x & 31;
    const int wave = threadIdx.x >> 5;
    const int tile = blockIdx.x * (blockDim.x >> 5) + wave;  // 64-row x 16-col tile

    const int ntilesN = N >> 4;
    const int tm = tile / ntilesN;
    const int tn = tile - tm * ntilesN;
    const int row_base = tm * (16 * MBLK);
    if (row_base >= M) return;                // wave-uniform guard

    const int half = lane >> 4;               // 0: K{0,1}, 1: K{2,3}
    const int l16  = lane & 15;

    const float* __restrict__ a_ptr = A + (size_t)(row_base + l16) * K + 2 * half;
    const float* __restrict__ w_ptr = W + (size_t)(2 * half) * N + tn * 16 + l16;

    v8f acc[MBLK] = {};

    // ---- prologue: fragments for k = 0
    v2f aF[MBLK], bF;
    bF.x = w_ptr[0];                          // row 2*half     (coalesced 64B)
    bF.y = w_ptr[N];                          // row 2*half + 1 (coalesced 64B)
    #pragma unroll
    for (int mi = 0; mi < MBLK; ++mi)
        aF[mi] = *reinterpret_cast<const v2f*>(a_ptr + (size_t)mi * 16 * K);

    // ---- pipelined main loop: load step k while computing step k-4
    #pragma unroll 2
    for (int k = 4; k <= K - 4; k += 4) {
        v2f aN[MBLK], bN;
        bN.x = w_ptr[(size_t)k * N];
        bN.y = w_ptr[(size_t)k * N + N];
        #pragma unroll
        for (int mi = 0; mi < MBLK; ++mi)
            aN[mi] = *reinterpret_cast<const v2f*>(
                         a_ptr + (size_t)mi * 16 * K + k);  // 8B-aligned b64

        #pragma unroll
        for (int mi = 0; mi < MBLK; ++mi)
            acc[mi] = __builtin_amdgcn_wmma_f32_16x16x4_f32(
                          false, aF[mi], false, bF, (short)0, acc[mi],
                          false, false);

        bF = bN;
        #pragma unroll
        for (int mi = 0; mi < MBLK; ++mi) aF[mi] = aN[mi];
    }

    // ---- epilogue: last k-step
    #pragma unroll
    for (int mi = 0; mi < MBLK; ++mi)
        acc[mi] = __builtin_amdgcn_wmma_f32_16x16x4_f32(
                      false, aF[mi], false, bF, (short)0, acc[mi],
                      false, false);

    const int   col = tn * 16 + l16;
    const float bv  = bias[col];
    #pragma unroll
    for (int mi = 0; mi < MBLK; ++mi) {
        const int row0 = row_base + mi * 16 + half * 8;
        #pragma unroll
        for (int i = 0; i < 8; ++i)
            out[(size_t)(row0 + i) * N + col] = acc[mi][i] + bv;
    }
}

// ---------------------------------------------------------------------------
// Fused softmax + bilinear sampling + weighted accumulation.
// One wave32 per (b, q, h); lane = channel d (hd = 32).
// Lanes 0..15 own sampling point j = lane (j = lvl*4 + p); softmax over the
// 16 logits is done with wave-wide shfl_xor butterflies (lanes >=16 feed
// identity elements). Per-point scalars are lane-broadcast so corner validity
// tests are wave-uniform (no divergence), and each corner fetch is a fully
// coalesced 128-byte read of the hd=32 contiguous channels (L2-resident).
// ---------------------------------------------------------------------------
__global__ __launch_bounds__(256)
void msdeform_sample(const float* __restrict__ value,    // (MROWS, 256)
                     const float* __restrict__ offs,     // (MROWS, 256)
                     const float* __restrict__ logits,   // (MROWS, 128)
                     const float* __restrict__ refpts,   // (B, NTOK, 4, 2)
                     float* __restrict__ headout) {      // (MROWS, 256)
    const int lane = threadIdx.x & 31;
    const int wave = threadIdx.x >> 5;
    const int idx  = blockIdx.x * (blockDim.x >> 5) + wave; // (b*NTOK+q)*8 + h
    if (idx >= MROWS * NHEADS) return;

    const int bq = idx >> 3;                 // b*NTOK + q
    const int h  = idx & 7;
    const int b  = bq / NTOK;

    // ---- lanes 0..15: load logit / offsets / reference for point j = lane
    float logit = -3.4e38f;
    float gx = 0.f, gy = 0.f;
    if (lane < 16) {
        const int j   = lane;
        const int lvl = j >> 2;
        logit = logits[(size_t)bq * 128 + h * 16 + j];
        const float ox = offs[(size_t)bq * 256 + h * 32 + j * 2 + 0];
        const float oy = offs[(size_t)bq * 256 + h * 32 + j * 2 + 1];
        const float rx = refpts[((size_t)bq * 4 + lvl) * 2 + 0];
        const float ry = refpts[((size_t)bq * 4 + lvl) * 2 + 1];
        const float Sl = (float)(128 >> lvl);   // square levels: H == W
        // grid_sample(align_corners=False):  g = loc*S - 0.5,  loc = ref + off/S
        gx = rx * Sl + ox - 0.5f;
        gy = ry * Sl + oy - 0.5f;
    }

    // ---- softmax over the 16 points (wave-wide butterflies)
    float m = logit;
    #pragma unroll
    for (int s = 16; s >= 1; s >>= 1) m = fmaxf(m, __shfl_xor(m, s, 32));
    float e = (lane < 16) ? __expf(logit - m) : 0.f;
    float sum = e;
    #pragma unroll
    for (int s = 16; s >= 1; s >>= 1) sum += __shfl_xor(sum, s, 32);
    const float wgt = e / sum;

    // ---- accumulate 16 bilinear samples; lane = channel
    const int    col   = h * HDIM + lane;     // channel column in (.,256)
    const size_t vbase = (size_t)b * NTOK;
    float acc = 0.f;

    #pragma unroll
    for (int j = 0; j < 16; ++j) {
        const float w  = __shfl(wgt, j, 32);
        const float px = __shfl(gx,  j, 32);
        const float py = __shfl(gy,  j, 32);
        const int lvl  = j >> 2;
        const int Sl   = 128 >> lvl;
        const int base = (lvl > 0) * 16384 + (lvl > 1) * 4096 + (lvl > 2) * 1024;

        const float x0f = floorf(px), y0f = floorf(py);
        const int   x0  = (int)x0f,   y0  = (int)y0f;
        const float fx  = px - x0f,   fy  = py - y0f;

        const float w00 = (1.f - fy) * (1.f - fx) * w;
        const float w01 = (1.f - fy) * fx * w;
        const float w10 = fy * (1.f - fx) * w;
        const float w11 = fy * fx * w;

        const bool vx0 = (unsigned)x0       < (unsigned)Sl;
        const bool vx1 = (unsigned)(x0 + 1) < (unsigned)Sl;
        const bool vy0 = (unsigned)y0       < (unsigned)Sl;
        const bool vy1 = (unsigned)(y0 + 1) < (unsigned)Sl;

        const size_t row = vbase + base + (size_t)y0 * Sl + x0;
        // wave-uniform conditions -> no divergence; 128B coalesced loads
        if (vy0 && vx0) acc += value[ row            * 256 + col] * w00;
        if (vy0 && vx1) acc += value[(row + 1)       * 256 + col] * w01;
        if (vy1 && vx0) acc += value[(row + Sl)      * 256 + col] * w10;
        if (vy1 && vx1) acc += value[(row + Sl + 1)  * 256 + col] * w11;
    }

    headout[(size_t)bq * 256 + col] = acc;
}

// ---------------------------------------------------------------------------
// Host-side launch
// ---------------------------------------------------------------------------
extern "C" void kernel_launch(void* const* d_in, const int* in_sizes, int n_in,
                              void* d_out, int out_size, void* d_ws, size_t ws_size,
                              hipStream_t stream) {
    const float* q      = (const float*)d_in[0];
    const float* ref    = (const float*)d_in[1];
    const float* v_in   = (const float*)d_in[2];
    const float* W_off  = (const float*)d_in[3];
    const float* b_off  = (const float*)d_in[4];
    const float* W_attn = (const float*)d_in[5];
    const float* b_attn = (const float*)d_in[6];
    const float* W_val  = (const float*)d_in[7];
    const float* b_val  = (const float*)d_in[8];
    const float* W_out  = (const float*)d_in[9];
    const float* b_out  = (const float*)d_in[10];
    float* out = (float*)d_out;

    // workspace layout (~156 MB total)
    float* ws_val  = (float*)d_ws;                    // MROWS*256
    float* ws_off  = ws_val  + (size_t)MROWS * 256;   // MROWS*256
    float* ws_attn = ws_off  + (size_t)MROWS * 256;   // MROWS*128
    float* ws_head = ws_attn + (size_t)MROWS * 128;   // MROWS*256

    const int threads = 256;                          // 8 waves
    const int wavesPerBlk = threads / 32;

    // 1) value projection: (MROWS,256) @ (256,256)
    {
        int tiles = (MROWS / (16 * MBLK)) * (CDIM / 16);   // 680*16
        gemm_wmma_f32<<<tiles / wavesPerBlk, threads, 0, stream>>>(
            v_in, W_val, b_val, ws_val, MROWS, CDIM, CDIM);
    }
    // 2) offset projection: (MROWS,256) @ (256,256)
    {
        int tiles = (MROWS / (16 * MBLK)) * (CDIM / 16);
        gemm_wmma_f32<<<tiles / wavesPerBlk, threads, 0, stream>>>(
            q, W_off, b_off, ws_off, MROWS, CDIM, CDIM);
    }
    // 3) attention logits: (MROWS,256) @ (256,128)
    {
        int tiles = (MROWS / (16 * MBLK)) * (128 / 16);    // 680*8
        gemm_wmma_f32<<<tiles / wavesPerBlk, threads, 0, stream>>>(
            q, W_attn, b_attn, ws_attn, MROWS, 128, CDIM);
    }
    // 4) fused softmax + bilinear sampling + weighted sum
    {
        int waves  = MROWS * NHEADS;                  // one wave per (b,q,h)
        int blocks = waves / wavesPerBlk;
        msdeform_sample<<<blocks, threads, 0, stream>>>(
            ws_val, ws_off, ws_attn, ref, ws_head);
    }
    // 5) output projection: (MROWS,256) @ (256,256) -> d_out
    {
        int tiles = (MROWS / (16 * MBLK)) * (CDIM / 16);
        gemm_wmma_f32<<<tiles / wavesPerBlk, threads, 0, stream>>>(
            ws_head, W_out, b_out, out, MROWS, CDIM, CDIM);
    }
}